// TransitionGNN_7473243095609
// MI455X (gfx1250) — compile-verified
//
#include <hip/hip_runtime.h>
#include <cstdint>
#include <cstddef>

// Problem constants from the reference: N=16384, K=32, N_OF=7, N_EF=1, H=1
#define NN   16384
#define KK   32
#define NOF  7
#define ROW_PAD 36              // floats per LDS row: 32 data + 4 pad -> 144B, 16B aligned
#define WAVES_PER_BLOCK 8

__device__ __forceinline__ float bcast_lane(float v, int lane) {
    // wave32 broadcast: v_readlane_b32 (uniform SGPR result, reused by all lanes)
    return __int_as_float(__builtin_amdgcn_readlane(__float_as_int(v), lane));
}

__device__ __forceinline__ float fast_sigmoid(float u) {
    // sigmoid(u) = rcp(1 + exp(-u)); v_exp_f32 + v_rcp_f32 both run on the
    // TRANS pipe and co-execute with VALU. Avoids the IEEE f32 div expansion.
    const float e = __expf(-u);
    return __builtin_amdgcn_rcpf(1.0f + e);
}

__global__ __launch_bounds__(WAVES_PER_BLOCK * 32)
void gnn_fused_kernel(const float* __restrict__ of,    // (N,K,7)
                      const float* __restrict__ ef,    // (N,K,K,1)
                      const float* __restrict__ act,   // (N,7)
                      const float* __restrict__ Wni,   // (14,1)
                      const float* __restrict__ bni,   // (1,)
                      const float* __restrict__ We,    // (3,1)
                      const float* __restrict__ be,    // (1,)
                      const float* __restrict__ Wn,    // (1,1)
                      const float* __restrict__ bn,    // (1,)
                      const float* __restrict__ Wef,   // (1,1)
                      const float* __restrict__ bef,   // (1,)
                      float* __restrict__ out)         // (N,K,K,1)
{
    __shared__ __align__(16) float lds_ef[WAVES_PER_BLOCK][KK * ROW_PAD];

    const int lane = threadIdx.x & 31;
    const int wave = threadIdx.x >> 5;
    // node index: wave-uniform by construction; readfirstlane makes it provably scalar
    const int n = __builtin_amdgcn_readfirstlane((int)(blockIdx.x * WAVES_PER_BLOCK + wave));
    if (n >= NN) return;

    // ------------------------------------------------------------------
    // Stage the 4KB edge tile ef[n,:,:] into LDS with async global->LDS.
    // Chunk c (0..255) = 16 bytes; global side is fully contiguous/coalesced,
    // LDS side lands in padded rows (ROW_PAD floats) to spread banks.
    // ------------------------------------------------------------------
    const float* efbase = ef + (size_t)n * (KK * KK);
    const unsigned lds_base = (unsigned)(uintptr_t)(&lds_ef[wave][0]);
#pragma unroll
    for (int t = 0; t < 8; ++t) {
        const int c = t * 32 + lane;                       // B128 chunk id
        const float* g = efbase + c * 4;                   // coalesced 16B/lane
        const unsigned l = lds_base
                         + (unsigned)((c >> 3) * (ROW_PAD * 4) + (c & 7) * 16);
        asm volatile("global_load_async_to_lds_b128 %0, %1, off"
                     :: "v"(l), "v"(g) : "memory");
    }

    // ------------------------------------------------------------------
    // Phase A (overlaps the async copy): lane k computes
    //   hn[k] = of[n,k,:] . Wni[0:7] + action[n,:] . Wni[7:14] + b_ni
    // ------------------------------------------------------------------
    const float w_e0 = We[0], w_e1 = We[1], w_e2 = We[2], b_e = be[0];
    const float w_n  = Wn[0], b_n  = bn[0];
    const float w_ef = Wef[0], b_ef = bef[0];

    float hn = bni[0];
    const float* ofp = of + ((size_t)n * KK + lane) * NOF;
    const float* ap  = act + (size_t)n * NOF;              // uniform -> scalar loads
#pragma unroll
    for (int f = 0; f < NOF; ++f) hn = fmaf(ofp[f], Wni[f], hn);
#pragma unroll
    for (int f = 0; f < NOF; ++f) hn = fmaf(ap[f], Wni[NOF + f], hn);

    // wait for the tile, then pull my row (i = lane) into registers once;
    // it is reused by both edge passes.
    asm volatile("s_wait_asynccnt 0" ::: "memory");
    float er[KK];
    const float* lrow = &lds_ef[wave][lane * ROW_PAD];
#pragma unroll
    for (int t = 0; t < 8; ++t) {
        const float4 v = *(const float4*)(lrow + t * 4);   // ds_load_b128, 16B aligned
        er[t * 4 + 0] = v.x; er[t * 4 + 1] = v.y;
        er[t * 4 + 2] = v.z; er[t * 4 + 3] = v.w;
    }

    // ------------------------------------------------------------------
    // Phase B: he_sum[i] = sum_j relu(hn[i]*w0 + hn[j]*w1 + ef[i,j]*w2 + be)
    //          hn2[i]   = relu(he_sum[i]*w_n + b_n)
    // (hn[i]*w0 + be is loop-invariant per lane -> hoisted)
    // ------------------------------------------------------------------
    const float a_i = fmaf(hn, w_e0, b_e);
    float hsum = 0.0f;
#pragma unroll
    for (int j = 0; j < KK; ++j) {
        const float hj = bcast_lane(hn, j);
        const float v  = fmaf(hj, w_e1, fmaf(er[j], w_e2, a_i));
        hsum += fmaxf(v, 0.0f);
    }
    const float hn2 = fmaxf(fmaf(hsum, w_n, b_n), 0.0f);

    // ------------------------------------------------------------------
    // Phase C: second edge pass + readout + sigmoid, write row i = lane.
    // ------------------------------------------------------------------
    const float a2_i = fmaf(hn2, w_e0, b_e);
    float orow[KK];
#pragma unroll
    for (int j = 0; j < KK; ++j) {
        const float hj = bcast_lane(hn2, j);
        const float v  = fmaxf(fmaf(hj, w_e1, fmaf(er[j], w_e2, a2_i)), 0.0f);
        const float u  = fmaxf(fmaf(v, w_ef, b_ef), 0.0f);
        orow[j] = fast_sigmoid(u);
    }

    float* op = out + (size_t)n * (KK * KK) + (size_t)lane * KK; // 16B aligned
#pragma unroll
    for (int t = 0; t < 8; ++t) {
        float4 v;
        v.x = orow[t * 4 + 0]; v.y = orow[t * 4 + 1];
        v.z = orow[t * 4 + 2]; v.w = orow[t * 4 + 3];
        *(float4*)(op + t * 4) = v;                        // global_store_b128
    }
}

extern "C" void kernel_launch(void* const* d_in, const int* in_sizes, int n_in,
                              void* d_out, int out_size, void* d_ws, size_t ws_size,
                              hipStream_t stream) {
    (void)in_sizes; (void)n_in; (void)out_size; (void)d_ws; (void)ws_size;
    const float* of  = (const float*)d_in[0];
    const float* ef  = (const float*)d_in[1];
    const float* act = (const float*)d_in[2];
    const float* Wni = (const float*)d_in[3];
    const float* bni = (const float*)d_in[4];
    const float* We  = (const float*)d_in[5];
    const float* be  = (const float*)d_in[6];
    const float* Wn  = (const float*)d_in[7];
    const float* bn  = (const float*)d_in[8];
    const float* Wef = (const float*)d_in[9];
    const float* bef = (const float*)d_in[10];
    float* out = (float*)d_out;

    const dim3 grid((NN + WAVES_PER_BLOCK - 1) / WAVES_PER_BLOCK);
    const dim3 block(WAVES_PER_BLOCK * 32);
    gnn_fused_kernel<<<grid, block, 0, stream>>>(of, ef, act, Wni, bni, We, be,
                                                 Wn, bn, Wef, bef, out);
}